// GCN_20186346291609
// MI455X (gfx1250) — compile-verified
//
#include <hip/hip_runtime.h>
#include <hip/hip_bf16.h>
#include <math.h>

// GCN on MI455X (gfx1250, wave32).
//   layer: S = H @ W   (fp32 WMMA 16x16x4, 16 x (16*NT) tile per wave,
//                       W pre-transposed so both A and B frags are b64 loads
//                       with immediate-folded k offsets)
//          O = bias-init; scatter atomicAdd over edges (support rows in 192MB L2)
//   final: row softmax (one wave per row, 40 classes)

#define NFEAT 512
#define NHID  256
#define NCLASS 40

typedef __attribute__((ext_vector_type(2))) float v2f;
typedef __attribute__((ext_vector_type(8))) float v8f;

// ---------------------------------------------------------------------------
// WT[n*K + k] = W[k*Nout + n]  (tiny: <= 512KB, runs once per layer)
// ---------------------------------------------------------------------------
__global__ void gcn_transpose_w(const float* __restrict__ W,
                                float* __restrict__ WT,
                                unsigned total, int K, int Nout)
{
    unsigned i = blockIdx.x * blockDim.x + threadIdx.x;   // over K*Nout
    if (i >= total) return;
    unsigned k = i / (unsigned)Nout;
    unsigned n = i - k * (unsigned)Nout;
    WT[(size_t)n * K + k] = W[i];
}

// ---------------------------------------------------------------------------
// C[M x Nout] = A[M x K] @ B[K x Nout], with B supplied TRANSPOSED
// (BT[Nout x K]).  M % 16 == 0, K % 4 == 0.  Each wave computes a
// 16 x (16*NT) tile with NT accumulators of V_WMMA_F32_16X16X4_F32.
// Out-of-range columns are CLAMPED (their products land only in output
// columns never stored), so all K-loop loads are unconditional and EXEC
// stays all-1s (WMMA requirement), with zero saveexec in the hot loop.
// Both A and B fragments are contiguous float2 -> global_load_b64 with
// k*4 folding into the instruction immediate offset under unrolling.
// ---------------------------------------------------------------------------
template<int NT>
__global__ void gcn_wmma_gemm_f32(const float* __restrict__ A,
                                  const float* __restrict__ BT,
                                  float* __restrict__ C,
                                  int M, int K, int Nout, int ngroupsN)
{
    const int wave = blockIdx.x * (blockDim.x >> 5) + (threadIdx.x >> 5);
    const int lane = threadIdx.x & 31;
    const int tM = wave / ngroupsN;
    const int tG = wave - tM * ngroupsN;
    if (tM * 16 >= M) return;                 // wave-uniform exit

    const int lo = lane & 15;
    const int hi = lane >> 4;                 // 0 or 1 (half-wave)

    // A: lane holds one row; VGPRs walk K (K = k+2*hi, k+2*hi+1)
    const int arow = tM * 16 + lo;
    const float* Ap = A + (size_t)arow * K + 2 * hi;

    // B: lane holds one column of B == one ROW of BT; VGPRs walk K
    int  col[NT];
    bool valid[NT];
    const float* Bp[NT];
    #pragma unroll
    for (int t = 0; t < NT; ++t) {
        int c    = (tG * NT + t) * 16 + lo;
        valid[t] = (c < Nout);
        col[t]   = valid[t] ? c : (Nout - 1);     // clamp: stays in-bounds
        Bp[t]    = BT + (size_t)col[t] * K + 2 * hi;
    }

    v8f acc[NT] = {};

    #pragma unroll 4
    for (int k = 0; k < K; k += 4) {
        v2f a = *(const v2f*)(Ap + k);            // global_load_b64
        #pragma unroll
        for (int t = 0; t < NT; ++t) {
            v2f b = *(const v2f*)(Bp[t] + k);     // global_load_b64 (contig in K)
            acc[t] = __builtin_amdgcn_wmma_f32_16x16x4_f32(
                         /*neg_a=*/false, a, /*neg_b=*/false, b,
                         /*c_mod=*/(short)0, acc[t],
                         /*reuse_a=*/false, /*reuse_b=*/false);
        }
    }

    // C/D layout: VGPR r -> M = r + 8*hi, N = lane&15
    #pragma unroll
    for (int t = 0; t < NT; ++t) {
        if (valid[t]) {
            float* Cp = C + (size_t)(tM * 16 + 8 * hi) * Nout + col[t];
            #pragma unroll
            for (int r = 0; r < 8; ++r)
                Cp[(size_t)r * Nout] = acc[t][r];
        }
    }
}

// ---------------------------------------------------------------------------
// out[i*Nout + f] = bias[f]   (initializes accumulation target)
// ---------------------------------------------------------------------------
__global__ void gcn_bias_init(float* __restrict__ out,
                              const float* __restrict__ bias,
                              unsigned total, int Nout)
{
    unsigned i = blockIdx.x * blockDim.x + threadIdx.x;
    if (i < total) out[i] = bias[i % Nout];
}

// ---------------------------------------------------------------------------
// Edge scatter: out[row] += val * support[col], 4 features per thread.
// support (<=102.4MB) is L2-resident; atomics land in L2.
// ---------------------------------------------------------------------------
__global__ void gcn_spmm_scatter(const float* __restrict__ support,
                                 const int* __restrict__ erow,
                                 const int* __restrict__ ecol,
                                 const float* __restrict__ eval,
                                 float* __restrict__ out,
                                 unsigned total, unsigned chunks, int Nout)
{
    unsigned tid = blockIdx.x * blockDim.x + threadIdx.x;
    if (tid >= total) return;
    unsigned e = tid / chunks;
    unsigned c = tid - e * chunks;

    const int   r  = erow[e];
    const int   cl = ecol[e];
    const float v  = eval[e];

    const float4 s = *(const float4*)(support + (size_t)cl * Nout + c * 4u);
    float* o = out + (size_t)r * Nout + c * 4u;
    atomicAdd(o + 0, v * s.x);
    atomicAdd(o + 1, v * s.y);
    atomicAdd(o + 2, v * s.z);
    atomicAdd(o + 3, v * s.w);
}

// ---------------------------------------------------------------------------
// Row softmax, 40 classes, one wave32 per row (lane covers c and c+32).
// ---------------------------------------------------------------------------
__global__ void gcn_softmax40(const float* __restrict__ in,
                              float* __restrict__ out, int rows)
{
    const int wave = blockIdx.x * (blockDim.x >> 5) + (threadIdx.x >> 5);
    const int lane = threadIdx.x & 31;
    if (wave >= rows) return;

    const float* rp = in + (size_t)wave * NCLASS;
    const bool hiOk = (lane + 32) < NCLASS;          // lanes 0..7
    float x0 = rp[lane];                              // lane < 32 <= 40: valid
    float x1 = hiOk ? rp[lane + 32] : -INFINITY;

    float m = fmaxf(x0, x1);
    #pragma unroll
    for (int off = 16; off > 0; off >>= 1)
        m = fmaxf(m, __shfl_xor(m, off, 32));

    float e0 = expf(x0 - m);
    float e1 = hiOk ? expf(x1 - m) : 0.0f;
    float s = e0 + e1;
    #pragma unroll
    for (int off = 16; off > 0; off >>= 1)
        s += __shfl_xor(s, off, 32);

    const float inv = 1.0f / s;
    float* op = out + (size_t)wave * NCLASS;
    op[lane] = e0 * inv;
    if (hiOk) op[lane + 32] = e1 * inv;
}

// ---------------------------------------------------------------------------
extern "C" void kernel_launch(void* const* d_in, const int* in_sizes, int n_in,
                              void* d_out, int out_size, void* d_ws, size_t ws_size,
                              hipStream_t stream)
{
    const float* x    = (const float*)d_in[0];
    const float* W0   = (const float*)d_in[1];
    const float* b0   = (const float*)d_in[2];
    const float* W1   = (const float*)d_in[3];
    const float* b1   = (const float*)d_in[4];
    const float* W2   = (const float*)d_in[5];
    const float* b2   = (const float*)d_in[6];
    const int*   erow = (const int*)d_in[7];
    const int*   ecol = (const int*)d_in[8];
    const float* eval = (const float*)d_in[9];
    float*       outp = (float*)d_out;

    const int Nn = in_sizes[0] / NFEAT;   // 100000 nodes
    const int E  = in_sizes[7];           // 3.2M edges

    float* bufA = (float*)d_ws;                    // 100000*256 f32 = 102.4 MB
    float* bufB = bufA + (size_t)Nn * NHID;        // second 102.4 MB
    float* wT   = bufB + (size_t)Nn * NHID;        // 512*256 f32 = 512 KB scratch

    const dim3 blk(256);                           // 8 waves per block
    const int mtiles = (Nn + 15) / 16;             // 6250

    auto transpose = [&](const float* W, int K, int Nout) {
        const unsigned total = (unsigned)K * (unsigned)Nout;
        gcn_transpose_w<<<(total + 255) / 256, blk, 0, stream>>>(W, wT, total, K, Nout);
    };
    // NT=4 -> 16x64 tiles (Nout=256: 4 groups); NT=3 -> 16x48 tile (Nout=40: 1 group)
    auto gemm256 = [&](const float* A, float* Cp, int K) {
        const int ngroups = (NHID + 63) / 64;      // 4
        const int waves   = mtiles * ngroups;
        gcn_wmma_gemm_f32<4><<<(waves + 7) / 8, blk, 0, stream>>>(
            A, wT, Cp, Nn, K, NHID, ngroups);
    };
    auto gemm40 = [&](const float* A, float* Cp, int K) {
        const int waves = mtiles;                  // 48 >= 40 cols in one group
        gcn_wmma_gemm_f32<3><<<(waves + 7) / 8, blk, 0, stream>>>(
            A, wT, Cp, Nn, K, NCLASS, 1);
    };
    auto spmm = [&](const float* S, const float* bias, float* O, int Nout) {
        const unsigned total = (unsigned)Nn * (unsigned)Nout;
        gcn_bias_init<<<(total + 255) / 256, blk, 0, stream>>>(O, bias, total, Nout);
        const unsigned chunks = (unsigned)(Nout >> 2);         // float4 per thread
        const unsigned tot    = (unsigned)E * chunks;
        gcn_spmm_scatter<<<(tot + 255) / 256, blk, 0, stream>>>(
            S, erow, ecol, eval, O, tot, chunks, Nout);
    };

    // Layer 0: S0 = x @ W0 ; H0 = A*S0 + b0
    transpose(W0, NFEAT, NHID);
    gemm256(x, bufA, NFEAT);
    spmm(bufA, b0, bufB, NHID);
    // Layer 1: S1 = H0 @ W1 ; H1 = A*S1 + b1
    transpose(W1, NHID, NHID);
    gemm256(bufB, bufA, NHID);
    spmm(bufA, b1, bufB, NHID);
    // Layer 2: S2 = H1 @ W2 ; H2 = A*S2 + b2
    transpose(W2, NHID, NCLASS);
    gemm40(bufB, bufA, NHID);
    spmm(bufA, b2, bufB, NCLASS);
    // Softmax -> d_out
    gcn_softmax40<<<(Nn + 7) / 8, blk, 0, stream>>>(bufB, outp, Nn);
}